// BlockadePEPS_5686536700067
// MI455X (gfx1250) — compile-verified
//
#include <hip/hip_runtime.h>
#include <stdint.h>

// BlockadePEPS greedy independent-set sampler for MI455X (gfx1250).
// One wave32 per sample, lane == column. Row recurrence done as 32-bit
// mask math (closed-form greedy run-parity selection). u is streamed
// into LDS with the Tensor Data Mover (double buffered), outputs are
// packed b128 stores.

typedef uint32_t u32;
typedef uint64_t u64;

#define WAVES_PER_BLOCK 8
#define TPB (WAVES_PER_BLOCK * 32)
#define CHUNK WAVES_PER_BLOCK              // samples per chunk (1 per wave)
#define SAMPLE_FLOATS 1024                 // 32x32
#define CHUNK_FLOATS (CHUNK * SAMPLE_FLOATS)
#define CHUNK_BYTES (CHUNK_FLOATS * 4)

typedef __attribute__((ext_vector_type(4))) u32 v4u;
typedef __attribute__((ext_vector_type(8))) int v8i;
typedef __attribute__((ext_vector_type(4))) int v4i;

__device__ __forceinline__ u32 ballot32(bool p) {
#if __has_builtin(__builtin_amdgcn_ballot_w32)
  return __builtin_amdgcn_ballot_w32(p);
#else
  return (u32)__ballot(p);
#endif
}

// Greedy left-to-right selection X[c] = a[c] & ~X[c-1], closed form:
// within each maximal run of 1s in a, keep bits whose position parity
// matches the run start's parity.
__device__ __forceinline__ u32 greedy_row(u32 b, u32 P) {
  u32 a = b & ~P;                 // candidates: rand bit set, up-neighbor clear
  u32 starts = a & ~(a << 1);     // one start bit per run
  u32 sE = starts & 0x55555555u;  // runs starting at even columns
  u32 t = a + sE;                 // carry sweeps exactly those runs
  u32 mE = ((t ^ a ^ sE) | sE) & a;           // full marks of even-start runs
  return (mE & 0x55555555u) | ((a & ~mE) & 0xAAAAAAAAu);
}

// src: 1024 floats for one sample (LDS or global). dst: 1024 int32 out.
template <typename PT>
__device__ __forceinline__ void process_sample(PT src, int* __restrict__ dst,
                                               int lane) {
  u32 P = 0u;
#pragma unroll
  for (int g = 0; g < 8; ++g) {   // 8 groups of 4 rows
    u32 m0, m1, m2, m3;
    {
      float v = src[(g * 4 + 0) * 32 + lane];
      m0 = greedy_row(ballot32(v < 0.5f), P); P = m0;
    }
    {
      float v = src[(g * 4 + 1) * 32 + lane];
      m1 = greedy_row(ballot32(v < 0.5f), P); P = m1;
    }
    {
      float v = src[(g * 4 + 2) * 32 + lane];
      m2 = greedy_row(ballot32(v < 0.5f), P); P = m2;
    }
    {
      float v = src[(g * 4 + 3) * 32 + lane];
      m3 = greedy_row(ballot32(v < 0.5f), P); P = m3;
    }
    // Pack 4 rows (128 int32 = 512B) into one b128 store per lane.
    // Lane L covers flat ints [4L, 4L+3] of the group: row = L/8, col0 = 4*(L&7).
    u32 msel = (lane & 16) ? ((lane & 8) ? m3 : m2)
                           : ((lane & 8) ? m1 : m0);
    int c0 = (lane & 7) * 4;
    int4 o;
    o.x = (int)((msel >> (c0 + 0)) & 1u);
    o.y = (int)((msel >> (c0 + 1)) & 1u);
    o.z = (int)((msel >> (c0 + 2)) & 1u);
    o.w = (int)((msel >> (c0 + 3)) & 1u);
    *reinterpret_cast<int4*>(dst + g * 128 + lane * 4) = o;
  }
}

#if defined(__gfx1250__) && __has_builtin(__builtin_amdgcn_tensor_load_to_lds)
#define USE_TDM 1
// Issue one TDM load: 2-D tile (1024 x m) of f32, tensor stride 1024,
// from u[sample s0 ...] into LDS at byte offset lds_addr.
__device__ __forceinline__ void issue_tdm(const float* u, int S, int chunk,
                                          u32 lds_addr) {
  int s0 = chunk * CHUNK;
  int m = S - s0;
  if (m > CHUNK) m = CHUNK;
  u64 ga = (u64)(uintptr_t)(u + (size_t)s0 * SAMPLE_FLOATS);

  v4u g0;
  g0[0] = 1u;                                   // count=1, user mode
  g0[1] = lds_addr;                             // lds_addr [63:32]
  g0[2] = (u32)ga;                              // global_addr lo
  g0[3] = ((u32)(ga >> 32) & 0x01FFFFFFu)       // global_addr[56:32]
          | (2u << 30);                         // type=2 ("image")

  const u32 td0 = 1024u;                        // tensor_dim0 (elements)
  const u32 t0  = 1024u;                        // tile_dim0
  v8i g1;
  g1[0] = (int)(2u << 16);                      // wg_mask=0, data_size=4B
  g1[1] = (int)((td0 & 0xFFFFu) << 16);         // tensor_dim0[15:0] @63:48
  g1[2] = (int)((td0 >> 16) | (((u32)m & 0xFFFFu) << 16)); // dim0 hi | dim1 lo
  g1[3] = (int)(((u32)m >> 16) | (t0 << 16));   // dim1 hi | tile_dim0 @127:112
  g1[4] = (int)((u32)m & 0xFFFFu);              // tile_dim1=m, tile_dim2=0
  g1[5] = (int)1024;                            // tensor_dim0_stride lo
  g1[6] = 0;                                    // stride hi | dim1_stride lo
  g1[7] = 0;
  v4i gz4 = {0, 0, 0, 0};
  v8i gz8 = {0, 0, 0, 0, 0, 0, 0, 0};
  // clang-23 / therock-10.0 lane: 6-arg form (extra unused v8i group).
  __builtin_amdgcn_tensor_load_to_lds(g0, g1, gz4, gz4, gz8, 0);
}
#else
#define USE_TDM 0
#endif

__global__ __launch_bounds__(TPB) void blockade_is_kernel(
    const float* __restrict__ u, int* __restrict__ out, int S) {
  const int lane = (int)(threadIdx.x & 31u);
  const int wave = (int)(threadIdx.x >> 5);
  const int chunks = (S + CHUNK - 1) / CHUNK;
  const int step = (int)gridDim.x;

#if USE_TDM
  __shared__ __align__(16) float smem[2 * CHUNK_FLOATS];  // 64 KB double buffer
  const u32 ldsBase = (u32)(uintptr_t)(&smem[0]);  // flat low 32 = LDS offset

  int i = 0;
  for (int cur = (int)blockIdx.x; cur < chunks; cur += step, ++i) {
    if (threadIdx.x < 32u) {                     // wave 0 drives the TDM
      if (i == 0) issue_tdm(u, S, cur, ldsBase);
      int next = cur + step;
      if (next < chunks) {
        issue_tdm(u, S, next, ldsBase + (u32)(((i + 1) & 1) * CHUNK_BYTES));
        __builtin_amdgcn_s_wait_tensorcnt(1);    // current chunk has arrived
      } else {
        __builtin_amdgcn_s_wait_tensorcnt(0);
      }
    }
    __syncthreads();                             // publish LDS to all waves
    int sample = cur * CHUNK + wave;
    if (sample < S) {
      const float* fb = &smem[(i & 1) * CHUNK_FLOATS + wave * SAMPLE_FLOATS];
      process_sample(fb, out + (size_t)sample * SAMPLE_FLOATS, lane);
    }
    __syncthreads();                             // done reading before reuse
  }
#else
  for (int cur = (int)blockIdx.x; cur < chunks; cur += step) {
    int sample = cur * CHUNK + wave;
    if (sample < S) {
      process_sample(u + (size_t)sample * SAMPLE_FLOATS,
                     out + (size_t)sample * SAMPLE_FLOATS, lane);
    }
  }
#endif
}

extern "C" void kernel_launch(void* const* d_in, const int* in_sizes, int n_in,
                              void* d_out, int out_size, void* d_ws,
                              size_t ws_size, hipStream_t stream) {
  (void)n_in; (void)out_size; (void)d_ws; (void)ws_size;
  const float* u = (const float*)d_in[0];
  int* out = (int*)d_out;
  int S = in_sizes[0] / SAMPLE_FLOATS;           // 65536 samples of 32x32
  int chunks = (S + CHUNK - 1) / CHUNK;
  int blocks = chunks < 2048 ? chunks : 2048;    // grid-stride, >=4 chunks/blk
  if (blocks < 1) blocks = 1;
  blockade_is_kernel<<<dim3(blocks), dim3(TPB), 0, stream>>>(u, out, S);
}